// Generator_53652731462112
// MI455X (gfx1250) — compile-verified
//
#include <hip/hip_runtime.h>
#include <math.h>

// TRADE generator dims
#define S_  30
#define B_  16
#define L_  400
#define H_  400
#define V_  18000
#define T_  10
#define G_  3
#define SB_ (S_ * B_)

typedef __attribute__((ext_vector_type(2))) float v2f;
typedef __attribute__((ext_vector_type(8))) float v8f;

// ---------------------------------------------------------------------------
// Generic fp32 WMMA GEMM:  D[m,n] = sum_k A[m,k] * B[k,n]  (+ bias[n])
//   A: element A[m][k] at A[z*aBatch + m*lda + k]
//   wTrans=1:  B[k][n] = W[z*wBatch + n*ldw + k]   (A @ W^T, W stored [N,K])
//   wTrans=0:  B[k][n] = W[z*wBatch + k*ldw + n]   (A @ W,   W stored [K,N])
//   D: element D[m][n] at D[z*dBatch + m*ldd + n]
// One wave computes a 16x64 tile (4 n-subtiles sharing the A fragment).
// N must be a multiple of 16, K a multiple of 4, lda/ldw even.
// Ragged M and ragged N-tile edges are handled by CLAMPING the load indices
// (duplicate finite data, never stored) so the inner loop is branch-free and
// EXEC stays all-ones around every v_wmma.
// ---------------------------------------------------------------------------
__global__ void __launch_bounds__(128) wmma_gemm_f32(
    const float* __restrict__ A, int lda, long aBatch,
    const float* __restrict__ W, int ldw, long wBatch,
    float* __restrict__ D, int ldd, long dBatch,
    int M, int N, int K, int wTrans, const float* __restrict__ bias)
{
    const int lane = threadIdx.x & 31;
    const int wave = threadIdx.x >> 5;
    const int col  = lane & 15;     // N (and A-M) index within tile
    const int hi   = lane >> 4;     // lane half selects K offset / D row half
    const int khi  = hi ? 2 : 0;
    const int tilesN = N >> 4;
    const int tnBase = (blockIdx.x * 4 + wave) * 4;
    if (tnBase >= tilesN) return;
    const int nsub = (tilesN - tnBase) < 4 ? (tilesN - tnBase) : 4;
    const int tm = blockIdx.y;
    const int z  = blockIdx.z;

    const float* Ab = A + (long)z * aBatch;
    const float* Wb = W + (long)z * wBatch;
    float*       Db = D + (long)z * dBatch;

    // A fragment row, clamped for ragged M (clamped rows only feed unstored
    // D rows, so duplicate data is harmless and the load needs no guard).
    const int am  = tm * 16 + col;
    const int amc = (am < M) ? am : (M - 1);
    const float* Arow = Ab + (long)amc * lda + khi;

    // Per-subtile B column indices, clamped at the N edge (results discarded).
    int bn[4];
    const float* Wp[4];
#pragma unroll
    for (int j = 0; j < 4; ++j) {
        int tn = tnBase + j;
        if (tn >= tilesN) tn = tilesN - 1;
        bn[j] = tn * 16 + col;
        Wp[j] = Wb + (long)bn[j] * ldw + khi;   // used in wTrans mode
    }

    v8f zero = {};
    v8f acc[4];
#pragma unroll
    for (int j = 0; j < 4; ++j) acc[j] = zero;

    if (wTrans) {
#pragma unroll 2
        for (int k = 0; k < K; k += 4) {
            v2f a = *(const v2f*)(Arow + k);            // global_load_b64
#pragma unroll
            for (int j = 0; j < 4; ++j) {
                v2f b = *(const v2f*)(Wp[j] + k);       // global_load_b64
                acc[j] = __builtin_amdgcn_wmma_f32_16x16x4_f32(
                    false, a, false, b, (short)0, acc[j], false, false);
            }
        }
    } else {
#pragma unroll 2
        for (int k = 0; k < K; k += 4) {
            v2f a = *(const v2f*)(Arow + k);
            const float* W0 = Wb + (long)(k + khi) * ldw;
#pragma unroll
            for (int j = 0; j < 4; ++j) {
                v2f b;
                b.x = W0[bn[j]];
                b.y = W0[ldw + bn[j]];
                acc[j] = __builtin_amdgcn_wmma_f32_16x16x4_f32(
                    false, a, false, b, (short)0, acc[j], false, false);
            }
        }
    }

    // D 16x16 layout: VGPR r -> row (r + 8*hi), col = lane&15
#pragma unroll
    for (int j = 0; j < 4; ++j) {
        if (j < nsub) {
            const int cn = (tnBase + j) * 16 + col;
            const float bv = bias ? bias[cn] : 0.0f;
#pragma unroll
            for (int r8 = 0; r8 < 8; ++r8) {
                const int dm = tm * 16 + r8 + hi * 8;
                if (dm < M) Db[(long)dm * ldd + cn] = acc[j][r8] + bv;
            }
        }
    }
}

// ---------------------------------------------------------------------------
// h0[s*B+b, :] = encoded_hidden[b, :]
// ---------------------------------------------------------------------------
__global__ void hinit_kernel(float* __restrict__ h, const float* __restrict__ eh)
{
    int idx = blockIdx.x * blockDim.x + threadIdx.x;
    if (idx >= SB_ * H_) return;
    int j = idx % H_;
    int r = idx / H_;
    int b = r % B_;
    h[idx] = eh[b * H_ + j];
}

// ---------------------------------------------------------------------------
// x[t]: t==0 -> slot_embs[s];  t>0 -> emb[target[b,s,t-1]]
// ---------------------------------------------------------------------------
__global__ void xfill_kernel(float* __restrict__ x, const float* __restrict__ emb,
                             const float* __restrict__ slot, const int* __restrict__ target,
                             int t)
{
    int idx = blockIdx.x * blockDim.x + threadIdx.x;
    if (idx >= SB_ * H_) return;
    int j = idx % H_;
    int r = idx / H_;
    int b = r % B_;
    int s = r / B_;
    float v;
    if (t == 0) {
        v = slot[s * H_ + j];
    } else {
        int tok = target[(b * S_ + s) * T_ + (t - 1)];
        v = emb[(long)tok * H_ + j];
    }
    x[idx] = v;
}

// ---------------------------------------------------------------------------
// GRU pointwise: h <- (1-z)*n + z*h  (in-place; gi/gh already include biases)
// ---------------------------------------------------------------------------
__global__ void gru_kernel(float* __restrict__ h, const float* __restrict__ gi,
                           const float* __restrict__ gh)
{
    int idx = blockIdx.x * blockDim.x + threadIdx.x;
    if (idx >= SB_ * H_) return;
    int j = idx % H_;
    int r = idx / H_;
    const float* gir = gi + (long)r * 3 * H_;
    const float* ghr = gh + (long)r * 3 * H_;
    float rg = 1.0f / (1.0f + __expf(-(gir[j] + ghr[j])));
    float z  = 1.0f / (1.0f + __expf(-(gir[H_ + j] + ghr[H_ + j])));
    float n  = tanhf(gir[2 * H_ + j] + rg * ghr[2 * H_ + j]);
    h[idx] = (1.0f - z) * n + z * h[idx];
}

// ---------------------------------------------------------------------------
// Masked softmax over L (block of 128 = 4 wave32 per row); masked probs -> 0
// ---------------------------------------------------------------------------
__global__ void __launch_bounds__(128) softmaxl_kernel(float* __restrict__ prob,
                                                       const int* __restrict__ lens)
{
    const int r = blockIdx.x;
    const int b = r % B_;
    const int len = lens[b];
    float* row = prob + (long)r * L_;
    __shared__ float sh[4];

    float m = -3.0e38f;
    for (int l = threadIdx.x; l < L_; l += 128)
        if (l < len) m = fmaxf(m, row[l]);
    for (int o = 16; o > 0; o >>= 1) m = fmaxf(m, __shfl_xor(m, o, 32));
    if ((threadIdx.x & 31) == 0) sh[threadIdx.x >> 5] = m;
    __syncthreads();
    m = fmaxf(fmaxf(sh[0], sh[1]), fmaxf(sh[2], sh[3]));
    __syncthreads();

    float ssum = 0.0f;
    for (int l = threadIdx.x; l < L_; l += 128)
        if (l < len) ssum += __expf(row[l] - m);
    for (int o = 16; o > 0; o >>= 1) ssum += __shfl_xor(ssum, o, 32);
    if ((threadIdx.x & 31) == 0) sh[threadIdx.x >> 5] = ssum;
    __syncthreads();
    float inv = 1.0f / (sh[0] + sh[1] + sh[2] + sh[3]);

    for (int l = threadIdx.x; l < L_; l += 128)
        row[l] = (l < len) ? __expf(row[l] - m) * inv : 0.0f;
}

// ---------------------------------------------------------------------------
// Pointer-gen switch + (t==0) slot-gate output.
// switch[r] = sigmoid([h;ctx;x] . W_ratio + b);  gate = ctx @ W_gate + b
// ---------------------------------------------------------------------------
__global__ void __launch_bounds__(128) swgate_kernel(
    const float* __restrict__ h, const float* __restrict__ ctx, const float* __restrict__ x,
    const float* __restrict__ Wr, const float* __restrict__ Wrb,
    const float* __restrict__ Wg, const float* __restrict__ Wgb,
    float* __restrict__ sw, float* __restrict__ gateOut, int t)
{
    const int r = blockIdx.x;
    __shared__ float sh[16];
    const long base = (long)r * H_;
    float a = 0.0f, g0 = 0.0f, g1 = 0.0f, g2 = 0.0f;
    for (int k = threadIdx.x; k < H_; k += 128) {
        float hv = h[base + k], cv = ctx[base + k], xv = x[base + k];
        a += hv * Wr[k] + cv * Wr[H_ + k] + xv * Wr[2 * H_ + k];
        if (t == 0) {
            g0 += cv * Wg[k * G_ + 0];
            g1 += cv * Wg[k * G_ + 1];
            g2 += cv * Wg[k * G_ + 2];
        }
    }
    for (int o = 16; o > 0; o >>= 1) {
        a  += __shfl_xor(a,  o, 32);
        g0 += __shfl_xor(g0, o, 32);
        g1 += __shfl_xor(g1, o, 32);
        g2 += __shfl_xor(g2, o, 32);
    }
    const int w = threadIdx.x >> 5;
    if ((threadIdx.x & 31) == 0) {
        sh[w] = a; sh[4 + w] = g0; sh[8 + w] = g1; sh[12 + w] = g2;
    }
    __syncthreads();
    if (threadIdx.x == 0) {
        float A = sh[0] + sh[1] + sh[2] + sh[3] + Wrb[0];
        sw[r] = 1.0f / (1.0f + __expf(-A));
        if (t == 0) {
            gateOut[r * G_ + 0] = sh[4]  + sh[5]  + sh[6]  + sh[7]  + Wgb[0];
            gateOut[r * G_ + 1] = sh[8]  + sh[9]  + sh[10] + sh[11] + Wgb[1];
            gateOut[r * G_ + 2] = sh[12] + sh[13] + sh[14] + sh[15] + Wgb[2];
        }
    }
}

// ---------------------------------------------------------------------------
// Online softmax statistics over V per row (single pass over logits)
// ---------------------------------------------------------------------------
__global__ void __launch_bounds__(256) vstats_kernel(const float* __restrict__ logits,
                                                     float* __restrict__ rmax,
                                                     float* __restrict__ rsum)
{
    const int r = blockIdx.x;
    const float* row = logits + (long)r * V_;
    float m = -3.0e38f, s = 0.0f;
    for (int v = threadIdx.x; v < V_; v += 256) {
        float xv = row[v];
        float M = fmaxf(m, xv);
        s = s * __expf(m - M) + __expf(xv - M);
        m = M;
    }
    for (int o = 16; o > 0; o >>= 1) {
        float m2 = __shfl_xor(m, o, 32);
        float s2 = __shfl_xor(s, o, 32);
        float M = fmaxf(m, m2);
        s = s * __expf(m - M) + s2 * __expf(m2 - M);
        m = M;
    }
    __shared__ float shm[8], shs[8];
    if ((threadIdx.x & 31) == 0) { shm[threadIdx.x >> 5] = m; shs[threadIdx.x >> 5] = s; }
    __syncthreads();
    if (threadIdx.x == 0) {
        float M = shm[0], SS = shs[0];
        for (int i = 1; i < 8; ++i) {
            float M2 = fmaxf(M, shm[i]);
            SS = SS * __expf(M - M2) + shs[i] * __expf(shm[i] - M2);
            M = M2;
        }
        rmax[r] = M;
        rsum[r] = SS;
    }
}

// ---------------------------------------------------------------------------
// out[r, t, :] = switch[r] * softmax(logits[r, :])   (vectorized float4)
// ---------------------------------------------------------------------------
__global__ void vcombine_kernel(const float* __restrict__ logits,
                                const float* __restrict__ rmax,
                                const float* __restrict__ rsum,
                                const float* __restrict__ sw,
                                float* __restrict__ out, int t)
{
    const long total = (long)SB_ * (V_ / 4);
    long idx = (long)blockIdx.x * 256 + threadIdx.x;
    if (idx >= total) return;
    int r  = (int)(idx / (V_ / 4));
    int v4 = (int)(idx % (V_ / 4));
    float m = rmax[r];
    float scale = sw[r] / rsum[r];
    float4 xv = reinterpret_cast<const float4*>(logits + (long)r * V_)[v4];
    float4 o;
    o.x = __expf(xv.x - m) * scale;
    o.y = __expf(xv.y - m) * scale;
    o.z = __expf(xv.z - m) * scale;
    o.w = __expf(xv.w - m) * scale;
    reinterpret_cast<float4*>(out + ((long)r * T_ + t) * V_)[v4] = o;
}

// ---------------------------------------------------------------------------
// Pointer scatter: out[r, t, story[b,l]] += (1-switch[r]) * prob[r,l]
// Rows are disjoint across r; uses hardware global_atomic_add_f32.
// ---------------------------------------------------------------------------
__global__ void scatter_kernel(const float* __restrict__ prob, const float* __restrict__ sw,
                               const int* __restrict__ story, float* __restrict__ out, int t)
{
    int idx = blockIdx.x * 256 + threadIdx.x;
    if (idx >= SB_ * L_) return;
    int l = idx % L_;
    int r = idx / L_;
    int b = r % B_;
    float p = prob[idx];           // 0 for masked positions
    if (p != 0.0f) {
        int tok = story[b * L_ + l];
        unsafeAtomicAdd(out + ((long)r * T_ + t) * V_ + tok, (1.0f - sw[r]) * p);
    }
}

// ---------------------------------------------------------------------------
// Host launcher
// ---------------------------------------------------------------------------
extern "C" void kernel_launch(void* const* d_in, const int* in_sizes, int n_in,
                              void* d_out, int out_size, void* d_ws, size_t ws_size,
                              hipStream_t stream)
{
    (void)in_sizes; (void)n_in; (void)out_size; (void)ws_size;

    const float* encoded_hidden  = (const float*)d_in[0];   // [B,H]
    const float* encoded_outputs = (const float*)d_in[1];   // [B,L,H]
    const int*   encoded_lens    = (const int*)  d_in[2];   // [B]
    const int*   story           = (const int*)  d_in[3];   // [B,L]
    const int*   target          = (const int*)  d_in[4];   // [B,S,T]
    const float* emb             = (const float*)d_in[5];   // [V,H]
    const float* slot_embs       = (const float*)d_in[6];   // [S,H]
    const float* W_ih            = (const float*)d_in[7];   // [3H,H]
    const float* W_hh            = (const float*)d_in[8];   // [3H,H]
    const float* b_ih            = (const float*)d_in[9];   // [3H]
    const float* b_hh            = (const float*)d_in[10];  // [3H]
    const float* W_ratio_w       = (const float*)d_in[11];  // [3H,1] (flat 3H)
    const float* W_ratio_b       = (const float*)d_in[12];  // [1]
    const float* W_gate_w        = (const float*)d_in[13];  // [H,G]
    const float* W_gate_b        = (const float*)d_in[14];  // [G]

    float* out = (float*)d_out;                 // [S,B,T,V] then [S,B,G]
    float* ws  = (float*)d_ws;

    // workspace partition (floats): ~42.3 MB total
    float* h      = ws;                         // [SB,H]
    float* x      = h      + (long)SB_ * H_;    // [SB,H]
    float* gi     = x      + (long)SB_ * H_;    // [SB,3H]
    float* gh     = gi     + (long)SB_ * 3 * H_;// [SB,3H]
    float* ctx    = gh     + (long)SB_ * 3 * H_;// [SB,H]
    float* prob   = ctx    + (long)SB_ * H_;    // [SB,L] (scores -> probs)
    float* logits = prob   + (long)SB_ * L_;    // [SB,V]
    float* sw     = logits + (long)SB_ * V_;    // [SB]
    float* rmax   = sw     + SB_;               // [SB]
    float* rsum   = rmax   + SB_;               // [SB]

    float* gateOut = out + (long)SB_ * T_ * V_; // [S,B,G]

    const int SBH    = SB_ * H_;
    const int ewGrid = (SBH + 255) / 256;

    hinit_kernel<<<ewGrid, 256, 0, stream>>>(h, encoded_hidden);

    for (int t = 0; t < T_; ++t) {
        // decoder input
        xfill_kernel<<<ewGrid, 256, 0, stream>>>(x, emb, slot_embs, target, t);

        // GRU input/hidden GEMMs: [480,400] x [400,1200] (A @ W^T + bias)
        {
            dim3 g((75 + 15) / 16, 30, 1);
            wmma_gemm_f32<<<g, 128, 0, stream>>>(x, H_, 0, W_ih, H_, 0,
                                                 gi, 3 * H_, 0,
                                                 SB_, 3 * H_, H_, 1, b_ih);
            wmma_gemm_f32<<<g, 128, 0, stream>>>(h, H_, 0, W_hh, H_, 0,
                                                 gh, 3 * H_, 0,
                                                 SB_, 3 * H_, H_, 1, b_hh);
        }
        gru_kernel<<<ewGrid, 256, 0, stream>>>(h, gi, gh);  // h becomes h_new

        // attention scores: per-b [30,400] x [400,400]^T -> prob buffer
        {
            dim3 g((25 + 15) / 16, 2, B_);
            wmma_gemm_f32<<<g, 128, 0, stream>>>(h, B_ * H_, H_,
                                                 encoded_outputs, H_, (long)L_ * H_,
                                                 prob, B_ * L_, L_,
                                                 S_, L_, H_, 1, nullptr);
        }
        softmaxl_kernel<<<SB_, 128, 0, stream>>>(prob, encoded_lens);

        // context: per-b [30,400] x [400,400]
        {
            dim3 g((25 + 15) / 16, 2, B_);
            wmma_gemm_f32<<<g, 128, 0, stream>>>(prob, B_ * L_, L_,
                                                 encoded_outputs, H_, (long)L_ * H_,
                                                 ctx, B_ * H_, H_,
                                                 S_, H_, L_, 0, nullptr);
        }

        // pointer-gen switch (+ gate output at t==0)
        swgate_kernel<<<SB_, 128, 0, stream>>>(h, ctx, x, W_ratio_w, W_ratio_b,
                                               W_gate_w, W_gate_b, sw, gateOut, t);

        // vocab logits: [480,400] x [400,18000] (A @ emb^T) -- dominant GEMM
        {
            dim3 g((1125 + 15) / 16, 30, 1);
            wmma_gemm_f32<<<g, 128, 0, stream>>>(h, H_, 0, emb, H_, 0,
                                                 logits, V_, 0,
                                                 SB_, V_, H_, 1, nullptr);
        }
        vstats_kernel<<<SB_, 256, 0, stream>>>(logits, rmax, rsum);

        // final = switch * p_vocab  (then scatter adds (1-switch)*p_ctx)
        {
            long total = (long)SB_ * (V_ / 4);
            vcombine_kernel<<<(int)((total + 255) / 256), 256, 0, stream>>>(
                logits, rmax, rsum, sw, out, t);
        }
        scatter_kernel<<<(SB_ * L_ + 255) / 256, 256, 0, stream>>>(prob, sw, story, out, t);
    }
}